// ODEMultistep_28948079575543
// MI455X (gfx1250) — compile-verified
//
#include <hip/hip_runtime.h>

typedef __attribute__((ext_vector_type(2))) float v2f;
typedef __attribute__((ext_vector_type(8))) float v8f;

#define B_TOT   8192
#define TLEN    513          // stop + 1
#define STOPC   512
#define NSTEP   511          // t = 0..510 : one MLP eval per time index
#define HSTEP   1e-3f
#define INV2H   500.0f       // 1/(2h)
#define NH      200
#define NPAD    208
#define NTILE   13
#define NKB     50           // K-blocks of 4 in layer 2
#define W1S     10           // k-stride of W1 in LDS
#define HS      210          // k-stride of h1 buffer (bank-coprime, even for b64)

// float offsets into dynamic LDS
#define OFF_W2  0                              // fragment-major W2
#define W2FRAGS (NKB * NTILE * 64)             // 41600 floats
#define OFF_W1  (OFF_W2 + W2FRAGS)             // 41600
#define OFF_H   (OFF_W1 + NPAD * W1S)          // 43680 (even)
#define HBUF_SZ (16 * HS)                      // 3360 floats per wave
#define LDS_FLOATS (OFF_H + 4 * HBUF_SZ)       // 57120 floats
#define LDS_BYTES  (LDS_FLOATS * 4)            // 228480 B  (< 320KB/WGP)

__device__ __forceinline__ v8f wmma4(v2f a, v2f b, v8f c) {
  // D = A(16x4 f32) * B(4x16 f32) + C(16x16 f32)
  return __builtin_amdgcn_wmma_f32_16x16x4_f32(false, a, false, b, (short)0, c,
                                               false, false);
}
__device__ __forceinline__ v2f ld2(const float* p) { return *(const v2f*)p; }

__global__ void __launch_bounds__(128, 1)
ode_ms_kernel(const float* __restrict__ nr, const float* __restrict__ iv,
              const float* __restrict__ kg,
              const float* __restrict__ W1g, const float* __restrict__ b1g,
              const float* __restrict__ W2g, const float* __restrict__ b2g,
              const float* __restrict__ W3g, const float* __restrict__ b3g,
              const float* __restrict__ Wsg, const float* __restrict__ bsg,
              float* __restrict__ out) {
  extern __shared__ float lds[];
  const int tid = threadIdx.x;

  // ---- stage W2 into LDS in WMMA-fragment-major order -----------------------
  // fragment f = kb*NTILE + ti holds 64 floats; lane ln supplies
  //   {W2[k][n], W2[k+1][n]} with k = 4*kb + 2*(ln>>4), n = 16*ti + (ln&15)
  for (int i = tid; i < W2FRAGS; i += 128) {
    int frag = i >> 6, r = i & 63;
    int ln = r >> 1, j = r & 1;
    int kb = frag / NTILE, ti = frag % NTILE;
    int k = 4 * kb + 2 * (ln >> 4) + j;
    int n = 16 * ti + (ln & 15);
    lds[OFF_W2 + i] = (n < NH) ? W2g[k * NH + n] : 0.f;
  }
  // ---- W1 stored [n][k], stride 10: one b64 = B-fragment pair ---------------
  for (int i = tid; i < NPAD * W1S; i += 128) {
    int n = i / W1S, kk = i % W1S;
    lds[OFF_W1 + i] = (n < NH && kk < 8) ? W1g[kk * NH + n] : 0.f;
  }
  __syncthreads();

  const int lane = tid & 31;
  const int wave = tid >> 5;
  const int m    = lane & 15;   // batch-row-in-tile (A) / N-in-tile (B,C/D)
  const int hi   = lane >> 4;
  const int row  = blockIdx.x * 64 + wave * 16 + m;    // batch index
  float* hbuf = &lds[OFF_H + wave * HBUF_SZ];          // per-wave 16x210 buffer
  const float* w2f = &lds[OFF_W2 + lane * 2];          // this lane's frag slot

  // loop-invariant per-lane fragments of b1, b2, W3 (registers, not LDS)
  float b1f[NTILE], b2f[NTILE], w3f[NTILE];
#pragma unroll
  for (int ti = 0; ti < NTILE; ++ti) {
    const int n = 16 * ti + m;
    const bool ok = n < NH;
    b1f[ti] = ok ? b1g[n] : 0.f;
    b2f[ti] = ok ? b2g[n] : 0.f;
    w3f[ti] = ok ? W3g[n] : 0.f;
  }

  const float kval = kg[row];
  const float ws0 = Wsg[0], ws1 = Wsg[1], ws2 = Wsg[2];
  const float bsv = bsg[0], b3v = b3g[0];
  const float iv0 = iv[row * 3 + 0], iv1 = iv[row * 3 + 1], iv2 = iv[row * 3 + 2];

  float nr_prev = nr[row * TLEN + STOPC];  // (0-1) mod 513 = 512 wrap
  float nr_cur  = nr[row * TLEN + 0];
  float nr_next = nr[row * TLEN + 1];
  float ni = iv0;
  float dh1 = 0.f, dh2 = 0.f;              // d(t-1), d(t-2) ring
  if (hi == 0) {
    out[row * STOPC + 0] = iv0;
    out[row * STOPC + 1] = iv1;
    out[row * STOPC + 2] = iv2;
  }

#pragma unroll 1
  for (int t = 0; t < NSTEP; ++t) {
    // ---- features (A-matrix 16x8 built in registers via lane selects) ----
    float dnr = (nr_next - nr_prev) * INV2H;
    float f0 = nr_cur, f1 = ni, f2 = dnr, f3 = kval;
    float f4 = f0 * f0, f5 = f1 * f1, f6 = 2.f * f3 * f0, f7 = f0 * f2;
    v2f A0 = {hi ? f2 : f0, hi ? f3 : f1};   // K-block 0: K = {0,1}|{2,3}
    v2f A1 = {hi ? f6 : f4, hi ? f7 : f5};   // K-block 1: K = {4,5}|{6,7}

    // ---- layer 1: [16x8] @ [8x208] -> relu -> hbuf[row][k] ----
#pragma unroll
    for (int ti = 0; ti < NTILE; ++ti) {
      const int n = 16 * ti + m;
      v8f c = {0.f, 0.f, 0.f, 0.f, 0.f, 0.f, 0.f, 0.f};
      c = wmma4(A0, ld2(&lds[OFF_W1 + n * W1S + 2 * hi]), c);
      c = wmma4(A1, ld2(&lds[OFF_W1 + n * W1S + 4 + 2 * hi]), c);
      const float bias = b1f[ti];
#pragma unroll
      for (int v = 0; v < 8; ++v)
        hbuf[(v + 8 * hi) * HS + n] = fmaxf(c[v] + bias, 0.f);
    }
    asm volatile("" ::: "memory");   // keep stores before reads; DS is in-order

    // ---- layer 2: [16x200] @ [200x208], K streamed from LDS ----
    v8f c2[NTILE];
    {   // peeled kb = 0: inline C = 0, no accumulator zero-fill movs
      const v2f a = ld2(&hbuf[m * HS + 2 * hi]);
#pragma unroll
      for (int ti = 0; ti < NTILE; ++ti) {
        v8f z = {0.f, 0.f, 0.f, 0.f, 0.f, 0.f, 0.f, 0.f};
        c2[ti] = wmma4(a, ld2(&w2f[ti * 64]), z);
      }
    }
#pragma unroll 1
    for (int kb = 1; kb < NKB; ++kb) {
      const v2f a = ld2(&hbuf[m * HS + 4 * kb + 2 * hi]);     // A frag (b64)
      const float* wf = &w2f[kb * (NTILE * 64)];
#pragma unroll
      for (int ti = 0; ti < NTILE; ++ti)
        c2[ti] = wmma4(a, ld2(&wf[ti * 64]), c2[ti]);         // 256B-apart frags
    }
    asm volatile("" ::: "memory");

    // ---- layer 3: fused bias+relu, dot with W3 in registers ----
    float acc[8] = {0.f, 0.f, 0.f, 0.f, 0.f, 0.f, 0.f, 0.f};
#pragma unroll
    for (int ti = 0; ti < NTILE; ++ti) {
      const float bias2 = b2f[ti];
      const float w3v = w3f[ti];
#pragma unroll
      for (int v = 0; v < 8; ++v)
        acc[v] = fmaf(fmaxf(c2[ti][v] + bias2, 0.f), w3v, acc[v]);
    }
    // reduce over 16 lanes of each half; half h holds d[8h .. 8h+7] in acc[0..7]
#pragma unroll
    for (int v = 0; v < 8; ++v) {
      acc[v] += __shfl_xor(acc[v], 1, 32);
      acc[v] += __shfl_xor(acc[v], 2, 32);
      acc[v] += __shfl_xor(acc[v], 4, 32);
      acc[v] += __shfl_xor(acc[v], 8, 32);
    }
    float s = acc[0];
#pragma unroll
    for (int i2 = 1; i2 < 8; ++i2) s = ((m & 7) == i2) ? acc[i2] : s;
    float so = __shfl_xor(s, 16, 32);                 // other half's value
    float dt = (((m >> 3) == hi) ? s : so) + b3v;     // d(t) for this lane's row

    // ---- superposition + multistep update ----
    float this_d = fmaf(ws0, dh2, fmaf(ws1, dh1, fmaf(ws2, dt, bsv)));
    dh2 = dh1; dh1 = dt;
    if (t >= 2) {
      ni = fmaf(HSTEP, this_d, ni);                   // ni[t+1]
      if (hi == 0) out[row * STOPC + t + 1] = ni;
    } else {
      ni = (t == 0) ? iv1 : iv2;                      // warm-up window from iv
    }
    nr_prev = nr_cur; nr_cur = nr_next;
    nr_next = nr[row * TLEN + t + 2];                 // t+2 <= 512, in range
  }
}

extern "C" void kernel_launch(void* const* d_in, const int* in_sizes, int n_in,
                              void* d_out, int out_size, void* d_ws, size_t ws_size,
                              hipStream_t stream) {
  (void)in_sizes; (void)n_in; (void)d_ws; (void)ws_size; (void)out_size;
  const float* nr = (const float*)d_in[0];
  const float* iv = (const float*)d_in[1];
  const float* k  = (const float*)d_in[2];
  // d_in[3] = stop (scalar int) — baked as 512
  const float* W1 = (const float*)d_in[4];
  const float* b1 = (const float*)d_in[5];
  const float* W2 = (const float*)d_in[6];
  const float* b2 = (const float*)d_in[7];
  const float* W3 = (const float*)d_in[8];
  const float* b3 = (const float*)d_in[9];
  const float* Ws = (const float*)d_in[10];
  const float* bs = (const float*)d_in[11];
  float* out = (float*)d_out;

  hipFuncSetAttribute((const void*)ode_ms_kernel,
                      hipFuncAttributeMaxDynamicSharedMemorySize, LDS_BYTES);
  ode_ms_kernel<<<dim3(B_TOT / 64), dim3(128), LDS_BYTES, stream>>>(
      nr, iv, k, W1, b1, W2, b2, W3, b3, Ws, bs, out);
}